// HungarianMatcher_51324859187598
// MI455X (gfx1250) — compile-verified
//
#include <hip/hip_runtime.h>
#include <hip/hip_bf16.h>

#define BB 16
#define QQ 900
#define CC 92
#define TT 100
#define QTILES 57   // ceil(900/16)
#define TTILES 7    // ceil(100/16)
#define KSTEPS 23   // 92/4
#define INF_F 1e18f

typedef __attribute__((ext_vector_type(2))) float v2f;
typedef __attribute__((ext_vector_type(8))) float v8f;

// ---------------------------------------------------------------------------
// Kernel 1: row softmax of logits -> probs in workspace
// ---------------------------------------------------------------------------
__global__ void softmax_rows_kernel(const float* __restrict__ logits,
                                    float* __restrict__ probs) {
  int row = blockIdx.x * blockDim.x + threadIdx.x;
  if (row >= BB * QQ) return;
  const float* x = logits + (size_t)row * CC;
  float* p = probs + (size_t)row * CC;
  float mx = -INF_F;
  for (int c = 0; c < CC; ++c) mx = fmaxf(mx, x[c]);
  float s = 0.f;
  for (int c = 0; c < CC; ++c) {
    float e = __expf(x[c] - mx);
    p[c] = e;
    s += e;
  }
  float inv = 1.f / s;
  for (int c = 0; c < CC; ++c) p[c] *= inv;
}

// ---------------------------------------------------------------------------
// Kernel 2: cost matrix. One wave per 16x16 (q,t) tile per batch.
// class term via V_WMMA_F32_16X16X4_F32 (probs x onehot(labels)),
// box-L1 + GIoU elementwise in the same wave.
// A-row pointer is CLAMPED (not masked): an invalid A row only pollutes the
// D row of the same m, which is never stored -> unconditional b64 loads.
// Writes cost[b,q,t] to d_out and cost^T[b,t,q] to workspace.
// ---------------------------------------------------------------------------
__global__ void cost_tiles_kernel(const float* __restrict__ probs,
                                  const float4* __restrict__ pred_boxes,
                                  const int* __restrict__ tgt_labels,
                                  const float4* __restrict__ tgt_boxes,
                                  float* __restrict__ cost_out,
                                  float* __restrict__ costT) {
  int wave = (blockIdx.x * blockDim.x + threadIdx.x) >> 5;
  int lane = threadIdx.x & 31;
  if (wave >= BB * QTILES * TTILES) return;
  int b = wave / (QTILES * TTILES);
  int rem = wave % (QTILES * TTILES);
  int qt = rem / TTILES;
  int tt = rem % TTILES;
  int q0 = qt * 16, t0 = tt * 16;

  int nlow = lane & 15;        // t-column within tile / q-row for A
  int khalf = lane >> 4;       // which K-half this lane holds

  // B-matrix (one-hot of labels) data for this lane's column
  int tcol = t0 + nlow;
  int lbl = (tcol < TT) ? tgt_labels[b * TT + tcol] : -1;

  // A-matrix row for this lane (clamped; rows >= QQ produce unused D rows)
  int qrow = q0 + nlow;
  int qc = (qrow < QQ) ? qrow : (QQ - 1);
  const float2* arow2 =
      (const float2*)(probs + ((size_t)b * QQ + qc) * CC);

  v8f acc = {};
#pragma unroll
  for (int kk = 0; kk < KSTEPS; ++kk) {
    int ka = 4 * kk + 2 * khalf;
    float2 av = arow2[2 * kk + khalf];   // one global_load_b64 per K-step
    v2f a;  a.x = av.x; a.y = av.y;
    v2f bm; bm.x = (lbl == ka) ? 1.f : 0.f;
            bm.y = (lbl == ka + 1) ? 1.f : 0.f;
    acc = __builtin_amdgcn_wmma_f32_16x16x4_f32(
        false, a, false, bm, (short)0, acc, false, false);
  }

  // Elementwise box terms; C/D layout: element (m,n), m = r + 8*khalf,
  // n = lane&15, value in acc[r].
  float4 tb;
  bool tok = (tcol < TT);
  if (tok) tb = tgt_boxes[b * TT + tcol];
  float bx0 = 0.f, by0 = 0.f, bx1 = 0.f, by1 = 0.f, area_b = 0.f;
  if (tok) {
    bx0 = tb.x - 0.5f * tb.z; by0 = tb.y - 0.5f * tb.w;
    bx1 = tb.x + 0.5f * tb.z; by1 = tb.y + 0.5f * tb.w;
    area_b = (bx1 - bx0) * (by1 - by0);
  }

#pragma unroll
  for (int r = 0; r < 8; ++r) {
    int m = r + 8 * khalf;
    int q = q0 + m;
    if (q < QQ && tok) {
      float4 pbx = pred_boxes[(size_t)b * QQ + q];
      float l1 = fabsf(pbx.x - tb.x) + fabsf(pbx.y - tb.y) +
                 fabsf(pbx.z - tb.z) + fabsf(pbx.w - tb.w);
      float ax0 = pbx.x - 0.5f * pbx.z, ay0 = pbx.y - 0.5f * pbx.w;
      float ax1 = pbx.x + 0.5f * pbx.z, ay1 = pbx.y + 0.5f * pbx.w;
      float area_a = (ax1 - ax0) * (ay1 - ay0);
      float ltx = fmaxf(ax0, bx0), lty = fmaxf(ay0, by0);
      float rbx = fminf(ax1, bx1), rby = fminf(ay1, by1);
      float iw = fmaxf(rbx - ltx, 0.f), ih = fmaxf(rby - lty, 0.f);
      float inter = iw * ih;
      float uni = area_a + area_b - inter;
      float iou = inter / uni;
      float cx0 = fminf(ax0, bx0), cy0 = fminf(ay0, by0);
      float cx1 = fmaxf(ax1, bx1), cy1 = fmaxf(ay1, by1);
      float cw = fmaxf(cx1 - cx0, 0.f), ch = fmaxf(cy1 - cy0, 0.f);
      float area_c = cw * ch;
      float giou = iou - (area_c - uni) / area_c;
      float cval = 5.0f * l1 - acc[r] - 2.0f * giou;
      cost_out[((size_t)b * QQ + q) * TT + tcol] = cval;
      costT[((size_t)b * TT + tcol) * QQ + q] = cval;
    }
  }
}

// ---------------------------------------------------------------------------
// Kernel 3: Hungarian (JV) per batch. One 1024-thread block per batch.
// n = T = 100 rows, m = Q = 900 columns, cost^T rows are contiguous.
// ---------------------------------------------------------------------------
__global__ __launch_bounds__(1024)
void hungarian_kernel(const float* __restrict__ costT,
                      float* __restrict__ outQ,
                      float* __restrict__ outT) {
  const int n = TT;      // 100
  const int m = QQ;      // 900
  int b = blockIdx.x;
  int tid = threadIdx.x;
  int lane = tid & 31;
  int wid = tid >> 5;

  __shared__ float u[TT + 1];
  __shared__ float v[QQ + 1];
  __shared__ int p[QQ + 1];
  __shared__ float minv[QQ];
  __shared__ int way[QQ];
  __shared__ int used[QQ + 1];
  __shared__ int colRow[TT];
  __shared__ float red_val[32];
  __shared__ int red_idx[32];
  __shared__ int sh_j0;
  __shared__ int sh_j1;
  __shared__ float sh_delta;

  const float* cb = costT + (size_t)b * n * m;

  if (tid <= n) u[tid] = 0.f;
  if (tid <= m) { v[tid] = 0.f; p[tid] = n; }
  __syncthreads();

  for (int i = 0; i < n; ++i) {
    if (tid == 0) { p[m] = i; sh_j0 = m; }
    if (tid < m) { minv[tid] = INF_F; way[tid] = 0; used[tid] = 0; }
    if (tid == m) used[m] = 0;
    __syncthreads();

    while (true) {
      int j0 = sh_j0;
      if (p[j0] == n) break;              // uniform: p/sh_j0 stable here
      if (tid == j0) used[j0] = 1;
      __syncthreads();

      int i0 = p[j0];
      float ui0 = u[i0];
      float mval = INF_F;
      if (tid < m && !used[tid]) {
        float cur = cb[(size_t)i0 * m + tid] - ui0 - v[tid];
        if (cur < minv[tid]) { minv[tid] = cur; way[tid] = j0; }
        mval = minv[tid];
      }
      // block argmin (prefer lowest index on ties, like jnp.argmin)
      float bv = mval; int bi = tid;
      for (int off = 16; off > 0; off >>= 1) {
        float ov = __shfl_down(bv, off, 32);
        int oi = __shfl_down(bi, off, 32);
        if (ov < bv || (ov == bv && oi < bi)) { bv = ov; bi = oi; }
      }
      if (lane == 0) { red_val[wid] = bv; red_idx[wid] = bi; }
      __syncthreads();
      if (wid == 0) {
        bv = red_val[lane]; bi = red_idx[lane];
        for (int off = 16; off > 0; off >>= 1) {
          float ov = __shfl_down(bv, off, 32);
          int oi = __shfl_down(bi, off, 32);
          if (ov < bv || (ov == bv && oi < bi)) { bv = ov; bi = oi; }
        }
        if (lane == 0) { sh_delta = bv; sh_j1 = bi; }
      }
      __syncthreads();

      float delta = sh_delta;
      if (tid <= m) {
        if (used[tid]) {
          v[tid] -= delta;
          u[p[tid]] += delta;   // used columns map to distinct rows
        } else if (tid < m) {
          minv[tid] -= delta;
        }
      }
      if (tid == 0) sh_j0 = sh_j1;
      __syncthreads();
    }
    __syncthreads();
    if (tid == 0) {
      int j0 = sh_j0;
      while (j0 != m) { int j1 = way[j0]; p[j0] = p[j1]; j0 = j1; }
    }
    __syncthreads();
  }

  // col_for_row + rank sort (argsort of assigned columns, all distinct)
  if (tid < m) {
    int r = p[tid];
    if (r < n) colRow[r] = tid;
  }
  __syncthreads();
  if (tid < n) {
    int qv = colRow[tid];
    int rank = 0;
    for (int t2 = 0; t2 < n; ++t2) rank += (colRow[t2] < qv) ? 1 : 0;
    outQ[b * n + rank] = (float)qv;
    outT[b * n + rank] = (float)tid;
  }
}

// ---------------------------------------------------------------------------
extern "C" void kernel_launch(void* const* d_in, const int* in_sizes, int n_in,
                              void* d_out, int out_size, void* d_ws, size_t ws_size,
                              hipStream_t stream) {
  const float* pred_logits = (const float*)d_in[0];   // B*Q*C
  const float4* pred_boxes = (const float4*)d_in[1];  // B*Q*4
  const int* tgt_labels = (const int*)d_in[2];        // B*T
  const float4* tgt_boxes = (const float4*)d_in[3];   // B*T*4

  float* probs = (float*)d_ws;                              // B*Q*C floats
  float* costT = probs + (size_t)BB * QQ * CC;              // B*T*Q floats

  float* cost_out = (float*)d_out;                          // B*Q*T
  float* outQ = cost_out + (size_t)BB * QQ * TT;            // B*T
  float* outT = outQ + (size_t)BB * TT;                     // B*T

  {
    int rows = BB * QQ;
    int threads = 256;
    int blocks = (rows + threads - 1) / threads;
    softmax_rows_kernel<<<blocks, threads, 0, stream>>>(pred_logits, probs);
  }
  {
    int waves = BB * QTILES * TTILES;
    int threads = 256;                      // 8 waves per block
    int blocks = (waves * 32 + threads - 1) / threads;
    cost_tiles_kernel<<<blocks, threads, 0, stream>>>(
        probs, pred_boxes, tgt_labels, tgt_boxes, cost_out, costT);
  }
  {
    hungarian_kernel<<<BB, 1024, 0, stream>>>(costT, outQ, outT);
  }
}